// SpatialTransformLayer_50560355008971
// MI455X (gfx1250) — compile-verified
//
#include <hip/hip_runtime.h>

// SpatialTransformLayer scatter: x[512,64,640] f32 -> grid[512,110,640] f32
// via fixed channel->cell permutation. Zero FLOPs; pure bandwidth problem
// (~228 MB total traffic => ~9.8us floor at 23.3 TB/s on MI455X).
//
// Single-kernel design (no setup launch, no inter-kernel bubble):
//   - each block rebuilds the 110-entry inverse map (cell -> channel or -1)
//     in LDS from the 64-entry channel_idx (L2-hot, ~700B/block extra reads)
//   - mapped output rows stream through LDS via the CDNA5 async copy path
//     (global_load_async_to_lds_b128 / global_store_async_from_lds_b128,
//     ASYNCcnt-tracked, single s_wait_asynccnt per wave)
//   - unmapped rows are zero-filled with nontemporal B128 stores (no read,
//     no separate memset pass)

#define N_B   512
#define N_CH  64
#define N_T   640
#define N_G   110                      // 10*11 grid cells
#define ROWS_TOTAL      (N_B * N_G)    // 56320 output rows
#define ROWS_PER_WAVE   2
#define WAVES_PER_BLOCK 8
#define BLOCK_THREADS   (WAVES_PER_BLOCK * 32)
#define SEGS            5              // 5 * 32 lanes * 16B = 2560B = one row
#define ROW_BYTES       (N_T * 4)

typedef float v4f __attribute__((ext_vector_type(4)));

__global__ __launch_bounds__(BLOCK_THREADS)
void scatter_async_kernel(const float* __restrict__ x,
                          const int* __restrict__ channel_idx,
                          float* __restrict__ out) {
  // Staging buffers for the async global->LDS->global row copies, plus the
  // per-block inverse map. ~41.4 KB total => 7 blocks resident per 320KB WGP.
  __shared__ unsigned char smem[WAVES_PER_BLOCK * ROWS_PER_WAVE * ROW_BYTES];
  __shared__ int inv_s[N_G];

  // ---- Build inverse map in LDS (channel_idx entries are distinct) ----
  if (threadIdx.x < N_G) inv_s[threadIdx.x] = -1;
  __syncthreads();
  if (threadIdx.x < N_CH) inv_s[channel_idx[threadIdx.x]] = threadIdx.x;
  __syncthreads();

  const int lane      = threadIdx.x & 31;         // wave32
  const int waveInBlk = threadIdx.x >> 5;
  const int waveId    = blockIdx.x * WAVES_PER_BLOCK + waveInBlk;
  const int rowBase   = waveId * ROWS_PER_WAVE;

  // Low 32 bits of the flat shared address == wave-relative LDS byte offset
  // (CDNA5 aperture mapping: the LDS address is the low 32 bits).
  const unsigned ldsWaveBase =
      (unsigned)(uintptr_t)(&smem[waveInBlk * ROWS_PER_WAVE * ROW_BYTES]);

  int cc[ROWS_PER_WAVE];

  // Phase 1: issue all async global->LDS loads for mapped rows
  // (up to 5120 B in flight per wave before the single ASYNCcnt wait).
#pragma unroll
  for (int r = 0; r < ROWS_PER_WAVE; ++r) {
    const int row = rowBase + r;
    cc[r] = -1;
    if (row < ROWS_TOTAL) {
      const int g = row % N_G;     // grid cell (uniform per wave)
      const int b = row / N_G;     // batch index
      const int c = inv_s[g];      // source channel or -1 (uniform per wave)
      cc[r] = c;
      if (c >= 0) {
        const float*   src  = x + ((size_t)(b * N_CH + c) * N_T);
        const unsigned lrow = ldsWaveBase + (unsigned)(r * ROW_BYTES);
#pragma unroll
        for (int s = 0; s < SEGS; ++s) {
          unsigned long long ga =
              (unsigned long long)(uintptr_t)(src + s * 128 + lane * 4);
          unsigned la = lrow + (unsigned)(s * 512 + lane * 16);
          asm volatile("global_load_async_to_lds_b128 %0, %1, off"
                       :: "v"(la), "v"(ga)
                       : "memory");
        }
      }
    }
  }

  // Wait for this wave's async loads to land in LDS.
  asm volatile("s_wait_asynccnt 0" ::: "memory");

  // Phase 2: async LDS->global stores for mapped rows; NT zero stores for
  // unmapped rows. S_ENDPGM's implicit wait-idle covers store completion.
#pragma unroll
  for (int r = 0; r < ROWS_PER_WAVE; ++r) {
    const int row = rowBase + r;
    if (row >= ROWS_TOTAL) continue;
    float* dst = out + (size_t)row * N_T;
    if (cc[r] >= 0) {
      const unsigned lrow = ldsWaveBase + (unsigned)(r * ROW_BYTES);
#pragma unroll
      for (int s = 0; s < SEGS; ++s) {
        unsigned long long ga =
            (unsigned long long)(uintptr_t)(dst + s * 128 + lane * 4);
        unsigned la = lrow + (unsigned)(s * 512 + lane * 16);
        asm volatile("global_store_async_from_lds_b128 %0, %1, off"
                     :: "v"(ga), "v"(la)
                     : "memory");
      }
    } else {
      v4f z = {0.0f, 0.0f, 0.0f, 0.0f};
#pragma unroll
      for (int s = 0; s < SEGS; ++s) {
        __builtin_nontemporal_store(z, (v4f*)(dst + s * 128 + lane * 4));
      }
    }
  }
}

// ---------------------------------------------------------------------------
// Host launcher: single kernel, exact grid (56320 rows / 2 per wave / 8 waves
// per block = 3520 blocks).
// ---------------------------------------------------------------------------
extern "C" void kernel_launch(void* const* d_in, const int* in_sizes, int n_in,
                              void* d_out, int out_size, void* d_ws, size_t ws_size,
                              hipStream_t stream) {
  (void)in_sizes; (void)n_in; (void)out_size; (void)d_ws; (void)ws_size;

  const float* x           = (const float*)d_in[0];
  const int*   channel_idx = (const int*)d_in[1];
  float*       out         = (float*)d_out;

  const int waveTasks = (ROWS_TOTAL + ROWS_PER_WAVE - 1) / ROWS_PER_WAVE;     // 28160
  const int blocks    = (waveTasks + WAVES_PER_BLOCK - 1) / WAVES_PER_BLOCK;  // 3520
  hipLaunchKernelGGL(scatter_async_kernel, dim3(blocks), dim3(BLOCK_THREADS),
                     0, stream, x, channel_idx, out);
}